// ACOPFEmbedder_39694087749652
// MI455X (gfx1250) — compile-verified
//
#include <hip/hip_runtime.h>
#include <hip/hip_bf16.h>
#include <math.h>

#define Hh 128
#define Ll 2
#define Nn 256
#define Ee 10000
#define ETt 11
#define NCc 22  // Ll*ETt

typedef float v2f __attribute__((ext_vector_type(2)));
typedef float v4f __attribute__((ext_vector_type(4)));
typedef float v8f __attribute__((ext_vector_type(8)));

// EDGE_TYPES: (SB,PQ)(PV,PQ)(NB,PQ)(PQ,NB)(PQ,SB)(PQ,PV)(PV,NB)(NB,PV)(PV,PV)(PQ,PQ)(NB,NB)
// type ids: SB=0 PQ=1 PV=2 NB=3
__constant__ int d_src[ETt]  = {0,2,3,1,1,1,2,3,2,1,3};
__constant__ int d_dst[ETt]  = {1,1,1,3,0,2,3,2,2,1,3};
__constant__ int d_rank[ETt] = {0,1,2,0,0,0,1,1,2,3,2};
__constant__ int d_cnt[4]    = {1,4,3,3};
__constant__ int d_pbase[4]  = {0,20000,100000,160000};   // rows per type: 20000,80000,60000,60000

__device__ __forceinline__ const float* selX(int t, const float* a, const float* b,
                                             const float* c, const float* d) {
  return t==0 ? a : (t==1 ? b : (t==2 ? c : d));
}

// ---------------- init: attention accumulators + out = fcb ----------------
__global__ void k_init(unsigned* mEnc, int* first, float* denom, float* wsum,
                       float* out, const float* fb0, const float* fb1,
                       const float* fb2, const float* fb3) {
  int i = blockIdx.x * blockDim.x + threadIdx.x;
  if (i < NCc * Nn) {
    mEnc[i]  = 0u;          // encoded -inf lower bound
    first[i] = 0x7fffffff;
    denom[i] = 0.f;
    wsum[i]  = 0.f;
  }
  if (i < 2048) {
    int ty = i >> 9, r = i & 511;
    const float* fb = ty==0 ? fb0 : (ty==1 ? fb1 : (ty==2 ? fb2 : fb3));
    out[i] = fb[r];
  }
}

// ---------------- collapse H=128 weights into 23 scalars per (l,t) --------
__global__ void k_coef(const float* __restrict__ Wq, const float* __restrict__ Wk,
                       const float* __restrict__ Wv, const float* __restrict__ We,
                       const float* __restrict__ Ws,
                       const float* __restrict__ bq, const float* __restrict__ bk,
                       const float* __restrict__ bv, const float* __restrict__ bs,
                       float* __restrict__ coef) {
  int c = blockIdx.x;      // 0..21 == l*ET+t
  int h = threadIdx.x;     // 0..127
  const int wb = c * 2 * Hh;
  const int bb = c * Hh;
  float wq0=Wq[wb+h], wq1=Wq[wb+Hh+h];
  float wk0=Wk[wb+h], wk1=Wk[wb+Hh+h];
  float wv0=Wv[wb+h], wv1=Wv[wb+Hh+h];
  float we0=We[wb+h], we1=We[wb+Hh+h];
  float ws0=Ws[wb+h], ws1=Ws[wb+Hh+h];
  float vbq=bq[bb+h], vbk=bk[bb+h], vbv=bv[bb+h], vbs=bs[bb+h];
  float p[23] = { wq0*wk0, wq0*wk1, wq1*wk0, wq1*wk1,     // A = Wq Wk^T
                  wq0*we0, wq0*we1, wq1*we0, wq1*we1,     // C = Wq We^T
                  wq0*vbk, wq1*vbk,                       // qb = Wq bk
                  wk0*vbq, wk1*vbq,                       // kb = Wk bq
                  we0*vbq, we1*vbq,                       // eb = We bq
                  vbq*vbk,                                // bb
                  wv0, wv1, we0, we1, vbv,                // sum_h v pieces
                  ws0, ws1, vbs };                        // sum_h skip pieces
  __shared__ float sm[128 * 24];
  #pragma unroll
  for (int i = 0; i < 23; i++) sm[h*24 + i] = p[i];
  __syncthreads();
  if (h < 23) {
    float s = 0.f;
    for (int k = 0; k < 128; k++) s += sm[k*24 + h];
    coef[c*24 + h] = s;
  }
}

// ---------------- per-edge logit / sum_h(v) / sum_h(skip) + seg max/min ---
__global__ void k_edge(const float* xSB, const float* xPQ, const float* xPV, const float* xNB,
                       const int* __restrict__ eidx, const float* __restrict__ eattr,
                       const float* __restrict__ coef,
                       float* __restrict__ logit, float* __restrict__ sv,
                       float* __restrict__ skip, unsigned* mEnc, int* first) {
  int c = blockIdx.y;
  int e = blockIdx.x * blockDim.x + threadIdx.x;
  if (e >= Ee) return;
  int t = c % ETt;
  const float* cf = coef + c * 24;
  int s = eidx[t*2*Ee + e];
  int d = eidx[t*2*Ee + Ee + e];
  const float* xsp = selX(d_src[t], xSB, xPQ, xPV, xNB);
  const float* xtp = selX(d_dst[t], xSB, xPQ, xPV, xNB);
  float xs0 = xsp[s*4+2], xs1 = xsp[s*4+3];
  float xt0 = xtp[d*4+2], xt1 = xtp[d*4+3];
  float ea0 = eattr[(t*Ee+e)*2], ea1 = eattr[(t*Ee+e)*2+1];
  float lg = xt0*(cf[0]*xs0 + cf[1]*xs1) + xt1*(cf[2]*xs0 + cf[3]*xs1)
           + xt0*(cf[4]*ea0 + cf[5]*ea1) + xt1*(cf[6]*ea0 + cf[7]*ea1)
           + xt0*cf[8] + xt1*cf[9]
           + cf[10]*xs0 + cf[11]*xs1
           + cf[12]*ea0 + cf[13]*ea1 + cf[14];
  lg *= 0.08838834764831845f;  // 1/sqrt(128)
  float svv = xs0*cf[15] + xs1*cf[16] + ea0*cf[17] + ea1*cf[18] + cf[19];
  float sk  = xt0*cf[20] + xt1*cf[21] + cf[22];
  int idx = c*Ee + e;
  logit[idx] = lg; sv[idx] = svv; skip[idx] = sk;
  unsigned b = __float_as_uint(lg);
  unsigned u = (b & 0x80000000u) ? ~b : (b | 0x80000000u);  // monotone encode
  atomicMax(&mEnc[c*Nn + d], u);
  atomicMin(&first[c*Nn + d], e);
}

// ---------------- segment softmax accumulation ----------------------------
__global__ void k_soft(const int* __restrict__ eidx, const float* __restrict__ logit,
                       const float* __restrict__ sv, const unsigned* __restrict__ mEnc,
                       float* denom, float* wsum) {
  int c = blockIdx.y;
  int e = blockIdx.x * blockDim.x + threadIdx.x;
  if (e >= Ee) return;
  int t = c % ETt;
  int d = eidx[t*2*Ee + Ee + e];
  unsigned u = mEnc[c*Nn + d];
  unsigned b = (u & 0x80000000u) ? (u ^ 0x80000000u) : ~u;
  float m = __uint_as_float(b);
  int idx = c*Ee + e;
  float ex = __expf(logit[idx] - m);
  atomicAdd(&denom[c*Nn + d], ex);
  atomicAdd(&wsum[c*Nn + d], ex * sv[idx]);
}

// ---------------- write per-row P into concatenation layout ---------------
__global__ void k_writep(const int* __restrict__ eidx, const float* __restrict__ skip,
                         const int* __restrict__ first, const float* __restrict__ denom,
                         const float* __restrict__ wsum, float* __restrict__ Pcat) {
  int c = blockIdx.y;
  int e = blockIdx.x * blockDim.x + threadIdx.x;
  if (e >= Ee) return;
  int t = c % ETt, l = c / ETt;
  int d = eidx[t*2*Ee + Ee + e];
  int sidx = c*Nn + d;
  float P = skip[c*Ee + e];
  if (first[sidx] == e) P += wsum[sidx] / denom[sidx];
  int dt  = d_dst[t];
  int blk = l * d_cnt[dt] + d_rank[t];
  Pcat[d_pbase[dt] + blk*Ee + e] = P;
}

// ---------------- dominant mat-vec: y[r] += W[r,2j]*P[j], via WMMA --------
// A tile 16x4 f32 per the ISA layout: lane -> M=lane&15, K base = 2*(lane>>4).
// One aligned b128 load per lane per 4-col step covers both needed even cols.
__global__ __launch_bounds__(32)
void k_fc_wmma(const float* __restrict__ W0, const float* __restrict__ W1,
               const float* __restrict__ W2, const float* __restrict__ W3,
               const float* __restrict__ Pcat, float* __restrict__ out) {
  int ty = blockIdx.z;
  const float* W  = ty==0 ? W0 : (ty==1 ? W1 : (ty==2 ? W2 : W3));
  const int Krows = ty==0 ? 20000 : (ty==1 ? 80000 : 60000);   // all multiples of 16
  const float* P  = Pcat + (ty==0 ? 0 : (ty==1 ? 20000 : (ty==2 ? 100000 : 160000)));
  int nch = (int)gridDim.x;
  int per = (Krows + nch - 1) / nch;
  per = (per + 15) & ~15;                       // multiple of 16 -> no tail, EXEC stays full
  int j0 = (int)blockIdx.x * per;
  if (j0 >= Krows) return;                      // uniform per wave
  int j1 = j0 + per; if (j1 > Krows) j1 = Krows;  // (j1-j0) % 16 == 0
  int lane = threadIdx.x;
  int M    = lane & 15;
  int half = lane >> 4;                         // 0 or 1
  size_t fan = (size_t)2 * (size_t)Krows;
  const float* row = W + (size_t)((int)blockIdx.y * 16 + M) * fan + 4 * (size_t)half;
  v8f acc0 = {0.f,0.f,0.f,0.f,0.f,0.f,0.f,0.f};
  v8f acc1 = acc0;
  for (int j = j0; j < j1; j += 16) {
    // prefetch this lane's row stream ~8KB ahead (near-cache)
    __builtin_prefetch((const void*)(row + 2*(size_t)j + 2048), 0, 3);
    #pragma unroll
    for (int u = 0; u < 4; u++) {
      int jj = j + 4*u;
      // A: elements 2*jj + 4*half + {0..3}, 16B aligned; need .x (K=2h) and .z (K=2h+1)
      v4f aw = *(const v4f*)(row + 2*(size_t)jj);
      // B: broadcast P[jj..jj+3], 16B aligned; half 0 -> K rows {0,2}, half 1 -> {1,3}
      v4f pw = *(const v4f*)(P + jj);
      v2f a, b;
      a.x = aw.x; a.y = aw.z;
      b.x = half ? pw.y : pw.x;
      b.y = half ? pw.w : pw.z;
      if (u & 1)
        acc1 = __builtin_amdgcn_wmma_f32_16x16x4_f32(false, a, false, b, (short)0, acc1,
                                                     false, false);
      else
        acc0 = __builtin_amdgcn_wmma_f32_16x16x4_f32(false, a, false, b, (short)0, acc0,
                                                     false, false);
    }
  }
  v8f acc = acc0 + acc1;
  // D layout: VGPR r, lanes0-15 -> M=r (N=lane), lanes16-31 -> M=r+8; all N cols equal
  int r0 = (int)blockIdx.y * 16;
  if (lane == 0) {
    #pragma unroll
    for (int i = 0; i < 8; i++) atomicAdd(&out[ty*512 + r0 + i], acc[i]);
  } else if (lane == 16) {
    #pragma unroll
    for (int i = 0; i < 8; i++) atomicAdd(&out[ty*512 + r0 + 8 + i], acc[i]);
  }
}

extern "C" void kernel_launch(void* const* d_in, const int* in_sizes, int n_in,
                              void* d_out, int out_size, void* d_ws, size_t ws_size,
                              hipStream_t stream) {
  (void)in_sizes; (void)n_in; (void)out_size; (void)ws_size;
  const float* xSB   = (const float*)d_in[0];
  const float* xPQ   = (const float*)d_in[1];
  const float* xPV   = (const float*)d_in[2];
  const float* xNB   = (const float*)d_in[3];
  const int*   eidx  = (const int*)  d_in[4];
  const float* eattr = (const float*)d_in[5];
  const float* Wq    = (const float*)d_in[6];
  const float* Wk    = (const float*)d_in[7];
  const float* Wv    = (const float*)d_in[8];
  const float* We    = (const float*)d_in[9];
  const float* Wsk   = (const float*)d_in[10];
  const float* bq    = (const float*)d_in[11];
  const float* bk    = (const float*)d_in[12];
  const float* bv    = (const float*)d_in[13];
  const float* bsk   = (const float*)d_in[14];
  const float* fcW_SB = (const float*)d_in[15];
  const float* fcb_SB = (const float*)d_in[16];
  const float* fcW_PQ = (const float*)d_in[17];
  const float* fcb_PQ = (const float*)d_in[18];
  const float* fcW_PV = (const float*)d_in[19];
  const float* fcb_PV = (const float*)d_in[20];
  const float* fcW_NB = (const float*)d_in[21];
  const float* fcb_NB = (const float*)d_in[22];
  float* out = (float*)d_out;

  // workspace layout (floats)
  float* wsf   = (float*)d_ws;
  float* coef  = wsf;                         // 528 (padded to 1024)
  float* logit = wsf + 1024;                  // 220000
  float* sv    = logit + NCc*Ee;              // 220000
  float* skip  = sv + NCc*Ee;                 // 220000
  unsigned* mEnc = (unsigned*)(skip + NCc*Ee);
  int*   first = (int*)(mEnc + NCc*Nn);
  float* denom = (float*)(first + NCc*Nn);
  float* wsum  = denom + NCc*Nn;
  float* Pcat  = wsum + NCc*Nn;               // 220000

  k_init<<<dim3((NCc*Nn + 255)/256), 256, 0, stream>>>(mEnc, first, denom, wsum,
                                                       out, fcb_SB, fcb_PQ, fcb_PV, fcb_NB);
  k_coef<<<dim3(NCc), 128, 0, stream>>>(Wq, Wk, Wv, We, Wsk, bq, bk, bv, bsk, coef);
  dim3 ge((Ee + 255)/256, NCc);
  k_edge<<<ge, 256, 0, stream>>>(xSB, xPQ, xPV, xNB, eidx, eattr, coef,
                                 logit, sv, skip, mEnc, first);
  k_soft<<<ge, 256, 0, stream>>>(eidx, logit, sv, mEnc, denom, wsum);
  k_writep<<<ge, 256, 0, stream>>>(eidx, skip, first, denom, wsum, Pcat);
  k_fc_wmma<<<dim3(64, 32, 4), 32, 0, stream>>>(fcW_SB, fcW_PQ, fcW_PV, fcW_NB, Pcat, out);
}